// ClfGCNGraph_69784628626008
// MI455X (gfx1250) — compile-verified
//
#include <hip/hip_runtime.h>
#include <hip/hip_bf16.h>
#include <math.h>

// Problem constants (match reference)
#define N_NODES  100000
#define N_EDGES  1600000
#define DIM      128
#define N_GRAPHS 100

typedef __attribute__((ext_vector_type(2))) float v2f;
typedef __attribute__((ext_vector_type(8))) float v8f;

// Native f32 global atomic add (global_atomic_add_f32, no CAS loop)
__device__ __forceinline__ void atomAddF(float* p, float v) {
    unsafeAtomicAdd(p, v);
}

// ---------------------------------------------------------------------------
// Kernel 0: zero the workspace (deterministic re-init every call)
// ---------------------------------------------------------------------------
__global__ void zero_ws_kernel(float* __restrict__ ws, long total) {
    long i = (long)blockIdx.x * blockDim.x + threadIdx.x;
    long stride = (long)gridDim.x * blockDim.x;
    for (; i < total; i += stride) ws[i] = 0.0f;
}

// ---------------------------------------------------------------------------
// Kernel 1: degree histograms via float atomics (L2-resident, 0.8 MB)
// ---------------------------------------------------------------------------
__global__ __launch_bounds__(256) void degree_kernel(
    const int* __restrict__ src, const int* __restrict__ dst,
    float* __restrict__ outdeg, float* __restrict__ indeg, int nE) {
    int e = blockIdx.x * 256 + threadIdx.x;
    if (e < nE) {
        atomAddF(&outdeg[src[e]], 1.0f);
        atomAddF(&indeg[dst[e]],  1.0f);
    }
}

// ---------------------------------------------------------------------------
// Kernel 2: edge message scatter. One wave per edge; each lane moves float4.
// agg[dst] += feat[src] * rsqrt(max(outdeg[src],1)) * e_w
// features + agg both fit in 192MB L2 -> L2-bandwidth bound.
// ---------------------------------------------------------------------------
__global__ __launch_bounds__(256) void edge_scatter_kernel(
    const float* __restrict__ feat, const float* __restrict__ ew,
    const int* __restrict__ src, const int* __restrict__ dst,
    const float* __restrict__ outdeg, float* __restrict__ agg, int nE) {
    int e    = blockIdx.x * 8 + (threadIdx.x >> 5);
    int lane = threadIdx.x & 31;
    if (e >= nE) return;
    int s = src[e];
    int d = dst[e];
    float w = ew[e] * rsqrtf(fmaxf(outdeg[s], 1.0f));
    const float4* f4 = (const float4*)(feat + (size_t)s * DIM);
    float4 v = f4[lane];
    float* ap = agg + (size_t)d * DIM + lane * 4;
    atomAddF(ap + 0, v.x * w);
    atomAddF(ap + 1, v.y * w);
    atomAddF(ap + 2, v.z * w);
    atomAddF(ap + 3, v.w * w);
}

// ---------------------------------------------------------------------------
// Kernel 3: node post-process + graph pooling.
// h = relu(agg * rsqrt(max(indeg,1)) + bias3); gsum[g] += h; gcnt[g] += 1
// One wave per node; pooled accumulators (51 KB) are L2-resident.
// ---------------------------------------------------------------------------
__global__ __launch_bounds__(256) void node_pool_kernel(
    const float* __restrict__ agg, const float* __restrict__ indeg,
    const float* __restrict__ bias3, const int* __restrict__ gids,
    float* __restrict__ gsum, float* __restrict__ gcnt, int nN) {
    int n    = blockIdx.x * 8 + (threadIdx.x >> 5);
    int lane = threadIdx.x & 31;
    if (n >= nN) return;
    float rs = rsqrtf(fmaxf(indeg[n], 1.0f));
    int g = gids[n];
    const float4* a4 = (const float4*)(agg + (size_t)n * DIM);
    const float4* b4 = (const float4*)bias3;
    float4 v  = a4[lane];
    float4 bb = b4[lane];
    float4 h;
    h.x = fmaxf(v.x * rs + bb.x, 0.0f);
    h.y = fmaxf(v.y * rs + bb.y, 0.0f);
    h.z = fmaxf(v.z * rs + bb.z, 0.0f);
    h.w = fmaxf(v.w * rs + bb.w, 0.0f);
    float* gp = gsum + g * DIM + lane * 4;
    atomAddF(gp + 0, h.x);
    atomAddF(gp + 1, h.y);
    atomAddF(gp + 2, h.z);
    atomAddF(gp + 3, h.w);
    if (lane == 0) atomAddF(&gcnt[g], 1.0f);
}

// ---------------------------------------------------------------------------
// Kernel 4: MLP head on [100,128] via V_WMMA_F32_16X16X4_F32.
// One wave per 16-graph row tile (7 blocks). All loads branch-free:
// padded rows are handled by clamped indices + multiply-by-zero, padded
// weights are staged into LDS once.
// A layout  (16x4):  lane<16 -> M=lane, K={0,1}; lane>=16 -> M=lane-16, K={2,3}
// B layout  (4x16):  lane<16 -> N=lane, K={0,1}; lane>=16 -> N=lane-16, K={2,3}
// C/D layout(16x16): VGPR r  -> row (lane<16 ? r : r+8), col = lane&15
// ---------------------------------------------------------------------------
#define TPAD (DIM + 4)   // row stride 132: 16B-aligned rows + bank rotation of 4

__global__ __launch_bounds__(32) void mlp_head_kernel(
    const float* __restrict__ gsum, const float* __restrict__ gcnt,
    const float* __restrict__ W1, const float* __restrict__ b1,
    const float* __restrict__ W2, const float* __restrict__ b2,
    const float* __restrict__ W3, const float* __restrict__ b3,
    float* __restrict__ out) {
    const int lane    = threadIdx.x;           // 0..31
    const int rowBase = blockIdx.x * 16;
    const int aRow    = lane & 15;             // M (and N for B frags)
    const int kOff    = (lane >> 4) * 2;       // 0 or 2

    __shared__ float sA[16][TPAD];             // hg tile (mean-normalized)
    __shared__ float sW2[16][16];              // zero-padded W2
    __shared__ float sW3[16];                  // zero-padded W3
    __shared__ float sH[16][17];               // inter-stage activations

    // ---- stage 0a: cooperative, branch-free tile load (2048 floats) ----
    // 16 float4 iterations; padded rows read clamped-valid memory, scaled by 0.
    for (int j = 0; j < 16; ++j) {
        int idx = j * 32 + lane;               // float4 index in tile
        int r   = idx >> 5;                    // tile row (32 float4 per row)
        int c4  = idx & 31;                    // float4 column
        int gg  = rowBase + r;
        int gc  = (gg < N_GRAPHS) ? gg : (N_GRAPHS - 1);
        float sc = (gg < N_GRAPHS) ? 1.0f / fmaxf(gcnt[gc], 1.0f) : 0.0f;
        float4 v = ((const float4*)(gsum + (size_t)gc * DIM))[c4];
        float4 w;
        w.x = v.x * sc; w.y = v.y * sc; w.z = v.z * sc; w.w = v.w * sc;
        ((float4*)&sA[r][0])[c4] = w;
    }
    // ---- stage 0b: padded weights into LDS (one-time setup) ----
    for (int j = 0; j < 8; ++j) {
        int idx = j * 32 + lane;               // 256 entries of sW2
        int k = idx >> 4, n = idx & 15;
        sW2[k][n] = (n < 8) ? W2[k * 8 + n] : 0.0f;
    }
    if (lane < 16) sW3[lane] = (lane < 8) ? W3[lane] : 0.0f;
    __syncthreads();

    // ---- stage 1: C1[16,16] = hg[16,128] @ W1[128,16] : 32 WMMA steps ----
    v8f acc = {};
    for (int k0 = 0; k0 < DIM; k0 += 4) {
        int kk = k0 + kOff;
        v2f a, b;
        a.x = sA[aRow][kk + 0];
        a.y = sA[aRow][kk + 1];
        b.x = W1[(kk + 0) * 16 + aRow];
        b.y = W1[(kk + 1) * 16 + aRow];
        acc = __builtin_amdgcn_wmma_f32_16x16x4_f32(
            false, a, false, b, (short)0, acc, false, false);
    }
    // H1 = relu(C1 + b1[col]) -> LDS (row-major)
    {
        const int col   = lane & 15;
        const int rBase = (lane >> 4) * 8;
        float bb = b1[col];
#pragma unroll
        for (int r = 0; r < 8; ++r)
            sH[rBase + r][col] = fmaxf(acc[r] + bb, 0.0f);
    }
    __syncthreads();

    // ---- stage 2: C2[16,16] = H1[16,16] @ W2p[16,16] : 4 WMMA steps ----
    v8f acc2 = {};
#pragma unroll
    for (int k0 = 0; k0 < 16; k0 += 4) {
        int kk = k0 + kOff;
        v2f a, b;
        a.x = sH[aRow][kk + 0];
        a.y = sH[aRow][kk + 1];
        b.x = sW2[kk + 0][aRow];
        b.y = sW2[kk + 1][aRow];
        acc2 = __builtin_amdgcn_wmma_f32_16x16x4_f32(
            false, a, false, b, (short)0, acc2, false, false);
    }
    __syncthreads();
    // H2 = relu(C2 + b2) on cols<8, zero elsewhere -> LDS
    {
        const int col   = lane & 15;
        const int rBase = (lane >> 4) * 8;
        float cm = (col < 8) ? 1.0f : 0.0f;    // arithmetic mask, no branch
        float bb = (col < 8) ? b2[col & 7] : 0.0f;
#pragma unroll
        for (int r = 0; r < 8; ++r)
            sH[rBase + r][col] = fmaxf(acc2[r] * cm + bb * cm, 0.0f) * cm;
    }
    __syncthreads();

    // ---- stage 3: C3[16,16] = H2[16,16] @ W3p[16,16] : 4 WMMA steps ----
    // W3 padded in LDS; only column 0 of B nonzero (mask via multiply).
    const float nMask = (aRow == 0) ? 1.0f : 0.0f;
    v8f acc3 = {};
#pragma unroll
    for (int k0 = 0; k0 < 16; k0 += 4) {
        int kk = k0 + kOff;
        v2f a, b;
        a.x = sH[aRow][kk + 0];
        a.y = sH[aRow][kk + 1];
        b.x = sW3[kk + 0] * nMask;
        b.y = sW3[kk + 1] * nMask;
        acc3 = __builtin_amdgcn_wmma_f32_16x16x4_f32(
            false, a, false, b, (short)0, acc3, false, false);
    }
    __syncthreads();
    {
        const int col   = lane & 15;
        const int rBase = (lane >> 4) * 8;
#pragma unroll
        for (int r = 0; r < 8; ++r)
            sH[rBase + r][col] = acc3[r];
    }
    __syncthreads();

    // out[g] = sigmoid(C3[row][0] + b3)
    if (lane < 16) {
        int gg = rowBase + lane;
        if (gg < N_GRAPHS) {
            float z = sH[lane][0] + b3[0];
            out[gg] = 1.0f / (1.0f + expf(-z));
        }
    }
}

// ---------------------------------------------------------------------------
// Launcher
// ---------------------------------------------------------------------------
extern "C" void kernel_launch(void* const* d_in, const int* in_sizes, int n_in,
                              void* d_out, int out_size, void* d_ws, size_t ws_size,
                              hipStream_t stream) {
    const float* feat  = (const float*)d_in[0];
    const float* ew    = (const float*)d_in[1];
    const int*   src   = (const int*)d_in[2];
    const int*   dst   = (const int*)d_in[3];
    const int*   gids  = (const int*)d_in[4];
    // bias1 (d_in[5]) and bias2 (d_in[6]) are dead in the reference: h is
    // overwritten by conv3, which consumes `features` directly.
    const float* bias3 = (const float*)d_in[7];
    const float* W1    = (const float*)d_in[8];
    const float* b1    = (const float*)d_in[9];
    const float* W2    = (const float*)d_in[10];
    const float* b2    = (const float*)d_in[11];
    const float* W3    = (const float*)d_in[12];
    const float* b3    = (const float*)d_in[13];
    float* out = (float*)d_out;

    // Workspace layout (floats): outdeg[N] | indeg[N] | gcnt[128] |
    //                            gsum[G*D] | agg[N*D]     (~52.1 MB total)
    float* ws     = (float*)d_ws;
    float* outdeg = ws;
    float* indeg  = ws + N_NODES;
    float* gcnt   = ws + 2 * N_NODES;
    float* gsum   = ws + 2 * N_NODES + 128;
    float* agg    = ws + 2 * N_NODES + 128 + N_GRAPHS * DIM;
    long   total  = 2L * N_NODES + 128 + N_GRAPHS * DIM + (long)N_NODES * DIM;

    zero_ws_kernel<<<2048, 256, 0, stream>>>(ws, total);
    degree_kernel<<<(N_EDGES + 255) / 256, 256, 0, stream>>>(
        src, dst, outdeg, indeg, N_EDGES);
    edge_scatter_kernel<<<N_EDGES / 8, 256, 0, stream>>>(
        feat, ew, src, dst, outdeg, agg, N_EDGES);
    node_pool_kernel<<<(N_NODES + 7) / 8, 256, 0, stream>>>(
        agg, indeg, bias3, gids, gsum, gcnt, N_NODES);
    mlp_head_kernel<<<(N_GRAPHS + 15) / 16, 32, 0, stream>>>(
        gsum, gcnt, W1, b1, W2, b2, W3, b3, out);
}